// LoraLinear_11914239279650
// MI455X (gfx1250) — compile-verified
//
#include <hip/hip_runtime.h>

typedef __attribute__((ext_vector_type(2))) float v2f;
typedef __attribute__((ext_vector_type(8))) float v8f;

#define IN_F   4096
#define OUT_F  4096
#define RANK   16
#define M_TOT  8192   // 4 * 2048

// ---------------------------------------------------------------------------
// Stage 1: t[m, r] = sum_k x[m, k] * A[r, k]     (t = x @ A^T)
// One 16-row M-tile per block; 4 waves each own a K-slice of 1024 and
// accumulate a 16x16 fp32 tile with V_WMMA_F32_16X16X4_F32, then reduce
// through LDS. x is streamed once -> non-temporal loads.
// ---------------------------------------------------------------------------
__global__ void __launch_bounds__(128)
lora_stage1_xAT(const float* __restrict__ x, const float* __restrict__ A,
                float* __restrict__ t)
{
    __shared__ float red[4 * 256];

    const int lane = threadIdx.x & 31;
    const int wave = threadIdx.x >> 5;        // 0..3 : K-slice owner
    const int m0   = blockIdx.x * 16;
    const int half = lane >> 4;               // 0: lanes 0-15, 1: lanes 16-31
    const int row  = lane & 15;

    // WMMA 32-bit operand layout: lane<16 holds (row, k0..k0+1),
    // lane>=16 holds (row, k0+2..k0+3)  => float2 load at +2*half.
    const float* xrow = x + (size_t)(m0 + row) * IN_F + 2 * half;  // A-operand
    const float* arow = A + (size_t)row * IN_F + 2 * half;         // B-operand (N=r)

    v8f acc = {};
    const int k_begin = wave * (IN_F / 4);    // 1024-wide K slice per wave
    const int k_end   = k_begin + (IN_F / 4);
    #pragma unroll 4
    for (int k = k_begin; k < k_end; k += 4) {
        v2f a = __builtin_nontemporal_load((const v2f*)(xrow + k)); // x: streamed once
        v2f b = *(const v2f*)(arow + k);                            // A: hot, 256 KB
        acc = __builtin_amdgcn_wmma_f32_16x16x4_f32(
                /*neg_a=*/false, a, /*neg_b=*/false, b,
                /*c_mod=*/(short)0, acc, /*reuse_a=*/false, /*reuse_b=*/false);
    }

    // Cross-wave reduction of the four partial 16x16 tiles via LDS.
    #pragma unroll
    for (int v = 0; v < 8; ++v)
        red[wave * 256 + v * 32 + lane] = acc[v];
    __syncthreads();

    if (wave == 0) {
        #pragma unroll
        for (int v = 0; v < 8; ++v) {
            float s = red[0 * 256 + v * 32 + lane]
                    + red[1 * 256 + v * 32 + lane]
                    + red[2 * 256 + v * 32 + lane]
                    + red[3 * 256 + v * 32 + lane];
            // C/D layout: VGPR v = row (v + 8*half), col = row-index r (N)
            const int m = m0 + v + 8 * half;
            t[(size_t)m * RANK + row] = s;
        }
    }
}

// ---------------------------------------------------------------------------
// Stage 2: out[m, n] = sum_r t[m, r] * B[n, r]   (out = t @ B^T)
// One 16x16 output tile per wave; R=16 => four K=4 WMMAs. t (512 KB) stays
// L2-resident across its 256x reuse; out is streamed -> non-temporal stores.
// ---------------------------------------------------------------------------
__global__ void __launch_bounds__(256)
lora_stage2_tBT(const float* __restrict__ t, const float* __restrict__ B,
                float* __restrict__ out)
{
    const int lane   = threadIdx.x & 31;
    const int waveId = blockIdx.x * 8 + (threadIdx.x >> 5);

    const int nTiles = OUT_F / 16;            // 256
    const int mTile  = waveId / nTiles;
    const int nTile  = waveId - mTile * nTiles;
    const int m0     = mTile * 16;
    const int n0     = nTile * 16;

    const int half = lane >> 4;
    const int idx  = lane & 15;               // M row (A-op) / N col (B-op)

    const float* trow = t + (size_t)(m0 + idx) * RANK + 2 * half;  // A-operand
    const float* brow = B + (size_t)(n0 + idx) * RANK + 2 * half;  // B-operand

    v8f acc = {};
    #pragma unroll
    for (int k = 0; k < RANK; k += 4) {
        v2f a = *(const v2f*)(trow + k);
        v2f b = *(const v2f*)(brow + k);
        acc = __builtin_amdgcn_wmma_f32_16x16x4_f32(
                false, a, false, b, (short)0, acc, false, false);
    }

    // Store: VGPR v holds (M = v + 8*half, N = n0 + idx).
    float* obase = out + (size_t)(m0 + 8 * half) * OUT_F + n0 + idx;
    #pragma unroll
    for (int v = 0; v < 8; ++v)
        __builtin_nontemporal_store(acc[v], obase + (size_t)v * OUT_F);
}

// ---------------------------------------------------------------------------
extern "C" void kernel_launch(void* const* d_in, const int* in_sizes, int n_in,
                              void* d_out, int out_size, void* d_ws, size_t ws_size,
                              hipStream_t stream) {
    const float* x = (const float*)d_in[0];   // [4, 2048, 4096]
    const float* B = (const float*)d_in[1];   // [4096, 16]
    const float* A = (const float*)d_in[2];   // [16, 4096]
    float* out = (float*)d_out;               // [4, 2048, 4096]
    float* t   = (float*)d_ws;                // [8192, 16] fp32 = 512 KB scratch

    // Stage 1: 512 M-tiles, 4 waves each (K-split), 128 threads/block.
    lora_stage1_xAT<<<M_TOT / 16, 128, 0, stream>>>(x, A, t);

    // Stage 2: (8192/16)*(4096/16) = 131072 tiles, 8 waves/block.
    const int tiles  = (M_TOT / 16) * (OUT_F / 16);
    lora_stage2_tBT<<<tiles / 8, 256, 0, stream>>>(t, B, out);
}